// PartSegmentation_26139170964326
// MI455X (gfx1250) — compile-verified
//
#include <hip/hip_runtime.h>
#include <hip/hip_bf16.h>
#include <math.h>

#define N_PTS 16384
#define M1C   4915
#define M2C   2458
#define KPTS  15
#define KNN   16
#define RAD1  0.2f
#define RAD2  0.4f
#define NCLS  50

typedef float v2f __attribute__((ext_vector_type(2)));
typedef float v8f __attribute__((ext_vector_type(8)));

static inline int cdiv_h(int a, int b) { return (a + b - 1) / b; }

__device__ __forceinline__ int imin_d(int a, int b) { return a < b ? a : b; }

// ---------------- min/max reduction + normalize ----------------
__global__ void minmax_partial(const float* __restrict__ x, int n, float* __restrict__ red) {
    __shared__ float smn[256], smx[256];
    int tid = threadIdx.x;
    float mn = 3.4e38f, mx = -3.4e38f;
    for (int i = blockIdx.x * blockDim.x + tid; i < n; i += gridDim.x * blockDim.x) {
        float v = x[i];
        mn = fminf(mn, v); mx = fmaxf(mx, v);
    }
    smn[tid] = mn; smx[tid] = mx;
    __syncthreads();
    for (int s = 128; s > 0; s >>= 1) {
        if (tid < s) { smn[tid] = fminf(smn[tid], smn[tid + s]); smx[tid] = fmaxf(smx[tid], smx[tid + s]); }
        __syncthreads();
    }
    if (tid == 0) { red[blockIdx.x] = smn[0]; red[64 + blockIdx.x] = smx[0]; }
}

__global__ void minmax_final(float* __restrict__ red) {
    if (threadIdx.x == 0 && blockIdx.x == 0) {
        float mn = 3.4e38f, mx = -3.4e38f;
        for (int i = 0; i < 64; i++) { mn = fminf(mn, red[i]); mx = fmaxf(mx, red[64 + i]); }
        red[128] = mn; red[129] = mx;
    }
}

__global__ void normalize_k(const float* __restrict__ pos, const float* __restrict__ red,
                            float* __restrict__ p, int n) {
    int i = blockIdx.x * blockDim.x + threadIdx.x;
    if (i >= n) return;
    float mn = red[128], mx = red[129];
    float c = (mx + mn) * 0.5f;
    float s = fabsf(mx - mn);
    p[i] = (pos[i] - c) / s;
}

// ---------------- brute-force kNN (top-KK, ascending d2, stable ties) ----------------
template <int KK>
__global__ void knn_kernel(const float* __restrict__ q, int Mq,
                           const float* __restrict__ src, int Ms,
                           int* __restrict__ oidx, float* __restrict__ od2) {
    int m = blockIdx.x * blockDim.x + threadIdx.x;
    if (m >= Mq) return;
    float qx = q[m * 3], qy = q[m * 3 + 1], qz = q[m * 3 + 2];
    float bd[KK]; int bi[KK];
#pragma unroll
    for (int i = 0; i < KK; i++) { bd[i] = 3.4e38f; bi[i] = 0; }
    for (int j = 0; j < Ms; j++) {
        float dx = qx - src[j * 3], dy = qy - src[j * 3 + 1], dz = qz - src[j * 3 + 2];
        float d2 = dx * dx + dy * dy + dz * dz;
        if (d2 < bd[KK - 1]) {
            float cd = d2; int ci = j;
#pragma unroll
            for (int i = 0; i < KK; i++) {
                bool sw = cd < bd[i];
                float td = bd[i]; int ti = bi[i];
                bd[i] = sw ? cd : td; bi[i] = sw ? ci : ti;
                cd = sw ? td : cd;   ci = sw ? ti : ci;
            }
        }
    }
#pragma unroll
    for (int i = 0; i < KK; i++) { oidx[m * KK + i] = bi[i]; od2[m * KK + i] = bd[i]; }
}

// ---------------- KPConv aggregation: g[m, k*C+c] = sum_n corr(m,n,k) * feat[idx(m,n), c]
// one wave per query point m; lane == channel c (clamped gather + value mask, no EXEC churn)
__global__ void kpconv_g(const float* __restrict__ qpos, int Mq,
                         const float* __restrict__ spos,
                         const float* __restrict__ feat, int C,
                         const int* __restrict__ nidx, const float* __restrict__ nd2,
                         const float* __restrict__ kp, float radius,
                         float* __restrict__ g) {
    int lane = threadIdx.x & 31;
    int m = blockIdx.x * (blockDim.x >> 5) + (threadIdx.x >> 5);
    if (m >= Mq) return;
    int lc = imin_d(lane, C - 1);               // clamped channel (always-legal load)
    bool lok = lane < C;
    float qx = qpos[m * 3], qy = qpos[m * 3 + 1], qz = qpos[m * 3 + 2];
    float rx[KNN], ry[KNN], rz[KNN], vl[KNN]; int ji[KNN];
    float r2 = radius * radius;
#pragma unroll
    for (int n = 0; n < KNN; n++) {
        int j = nidx[m * KNN + n]; ji[n] = j;
        rx[n] = spos[j * 3] - qx;
        ry[n] = spos[j * 3 + 1] - qy;
        rz[n] = spos[j * 3 + 2] - qz;
        vl[n] = (nd2[m * KNN + n] <= r2) ? 1.0f : 0.0f;
    }
    float inv_r = 1.0f / radius;
    for (int k = 0; k < KPTS; k++) {
        float kx = kp[k * 3], ky = kp[k * 3 + 1], kz = kp[k * 3 + 2];
        float acc = 0.0f;
#pragma unroll
        for (int n = 0; n < KNN; n++) {
            float dx = rx[n] - kx, dy = ry[n] - ky, dz = rz[n] - kz;
            float dk = sqrtf(dx * dx + dy * dy + dz * dz + 1e-12f);
            float corr = fmaxf(0.0f, 1.0f - dk * inv_r) * vl[n];
            float f = feat[(size_t)ji[n] * C + lc];
            acc += corr * (lok ? f : 0.0f);
        }
        if (lok) g[(size_t)m * (KPTS * C) + k * C + lane] = acc;
    }
}

// ---------------- fp32 WMMA GEMM: C = [relu](A(MxK) @ B(KxN) + bias) ----------------
// one 16x16 output tile per wave32; V_WMMA_F32_16X16X4_F32 stepping K by 4.
// Full tiles take an unguarded load path (pure load+wmma loop); edge tiles use
// clamped addresses with value masking (v_cndmask, no EXEC manipulation).
__global__ void gemm_wmma(const float* __restrict__ A, const float* __restrict__ B,
                          const float* __restrict__ bias, float* __restrict__ C,
                          int M, int Nc, int K, int relu) {
    int lane = threadIdx.x & 31;
    int wid = blockIdx.x * (blockDim.x >> 5) + (threadIdx.x >> 5);
    int tn = (Nc + 15) >> 4;
    int ttotal = ((M + 15) >> 4) * tn;
    if (wid >= ttotal) return;                 // wave-uniform exit: EXEC stays all-ones
    int tile_m = wid / tn, tile_n = wid % tn;
    int row0 = tile_m << 4, col0 = tile_n << 4;
    int l15 = lane & 15;
    int kb = (lane >> 4) << 1;                 // K sub-offset {0,2} per half-wave
    int am = row0 + l15;                       // A row handled by this lane
    int bn = col0 + l15;                       // B col handled by this lane

    bool fullT = (row0 + 16 <= M) && (col0 + 16 <= Nc);   // tile fully in-bounds (uniform)
    int Kmain = K & ~3;

    v8f acc = {0.f, 0.f, 0.f, 0.f, 0.f, 0.f, 0.f, 0.f};

    if (fullT) {
        const float* Ar = A + (size_t)am * K + kb;        // lane's row, K offset kb
        const float* Bc = B + bn;
        for (int k0 = 0; k0 < Kmain; k0 += 4) {
            v2f a, b;
            a.x = Ar[k0];                                  // contiguous pair -> b64 load
            a.y = Ar[k0 + 1];
            b.x = Bc[(size_t)(k0 + kb) * Nc];
            b.y = Bc[(size_t)(k0 + kb + 1) * Nc];
            acc = __builtin_amdgcn_wmma_f32_16x16x4_f32(false, a, false, b, (short)0, acc,
                                                        false, false);
        }
        if (Kmain < K) {                                   // K tail (K=45 case): mask values
            int ka = Kmain + kb, ka1 = ka + 1;
            int kac = imin_d(ka, K - 1), kac1 = imin_d(ka1, K - 1);
            v2f a, b;
            float a0 = A[(size_t)am * K + kac];
            float a1 = A[(size_t)am * K + kac1];
            float b0 = B[(size_t)kac * Nc + bn];
            float b1 = B[(size_t)kac1 * Nc + bn];
            a.x = (ka < K)  ? a0 : 0.0f;
            a.y = (ka1 < K) ? a1 : 0.0f;
            b.x = (ka < K)  ? b0 : 0.0f;
            b.y = (ka1 < K) ? b1 : 0.0f;
            acc = __builtin_amdgcn_wmma_f32_16x16x4_f32(false, a, false, b, (short)0, acc,
                                                        false, false);
        }
    } else {
        int amc = imin_d(am, M - 1);                       // clamped (always-legal) indices
        int bnc = imin_d(bn, Nc - 1);
        bool amok = am < M, bnok = bn < Nc;
        for (int k0 = 0; k0 < K; k0 += 4) {
            int ka = k0 + kb, ka1 = ka + 1;
            int kac = imin_d(ka, K - 1), kac1 = imin_d(ka1, K - 1);
            v2f a, b;
            float a0 = A[(size_t)amc * K + kac];
            float a1 = A[(size_t)amc * K + kac1];
            float b0 = B[(size_t)kac * Nc + bnc];
            float b1 = B[(size_t)kac1 * Nc + bnc];
            a.x = (amok && ka < K)  ? a0 : 0.0f;
            a.y = (amok && ka1 < K) ? a1 : 0.0f;
            b.x = (bnok && ka < K)  ? b0 : 0.0f;
            b.y = (bnok && ka1 < K) ? b1 : 0.0f;
            acc = __builtin_amdgcn_wmma_f32_16x16x4_f32(false, a, false, b, (short)0, acc,
                                                        false, false);
        }
    }

    int crow0 = row0 + ((lane >> 4) << 3);     // +8 for lanes 16..31
    int ccol = col0 + l15;
    if (ccol < Nc) {
        float bv = bias ? bias[ccol] : 0.0f;
#pragma unroll
        for (int r = 0; r < 8; r++) {
            int crow = crow0 + r;
            if (crow < M) {
                float v = acc[r] + bv;
                if (relu) v = fmaxf(v, 0.0f);
                C[(size_t)crow * Nc + ccol] = v;
            }
        }
    }
}

// ---------------- kNN(k=3) inverse-distance interpolation ----------------
__global__ void knn_interp3(const float* __restrict__ q, int Mq,
                            const float* __restrict__ src, int Ms,
                            const float* __restrict__ feat, int C,
                            float* __restrict__ out, int ostride, int ooff) {
    int m = blockIdx.x * blockDim.x + threadIdx.x;
    if (m >= Mq) return;
    float qx = q[m * 3], qy = q[m * 3 + 1], qz = q[m * 3 + 2];
    float bd[3] = {3.4e38f, 3.4e38f, 3.4e38f};
    int bi[3] = {0, 0, 0};
    for (int j = 0; j < Ms; j++) {
        float dx = qx - src[j * 3], dy = qy - src[j * 3 + 1], dz = qz - src[j * 3 + 2];
        float d2 = dx * dx + dy * dy + dz * dz;
        if (d2 < bd[2]) {
            float cd = d2; int ci = j;
#pragma unroll
            for (int i = 0; i < 3; i++) {
                bool sw = cd < bd[i];
                float td = bd[i]; int ti = bi[i];
                bd[i] = sw ? cd : td; bi[i] = sw ? ci : ti;
                cd = sw ? td : cd;   ci = sw ? ti : ci;
            }
        }
    }
    float w0 = 1.0f / fmaxf(bd[0], 1e-16f);
    float w1 = 1.0f / fmaxf(bd[1], 1e-16f);
    float w2 = 1.0f / fmaxf(bd[2], 1e-16f);
    float inv = 1.0f / (w0 + w1 + w2);
    w0 *= inv; w1 *= inv; w2 *= inv;
    const float* f0 = feat + (size_t)bi[0] * C;
    const float* f1 = feat + (size_t)bi[1] * C;
    const float* f2 = feat + (size_t)bi[2] * C;
    for (int c = 0; c < C; c++)
        out[(size_t)m * ostride + ooff + c] = w0 * f0[c] + w1 * f1[c] + w2 * f2[c];
}

// ---------------- copy x1 into columns [64:96) of u ----------------
__global__ void copy_block(const float* __restrict__ src, float* __restrict__ dst,
                           int M, int C, int dstride, int doff) {
    int i = blockIdx.x * blockDim.x + threadIdx.x;
    if (i >= M * C) return;
    int m = i / C, c = i % C;
    dst[(size_t)m * dstride + doff + c] = src[i];
}

// ---------------- row-wise log-softmax (in place) ----------------
__global__ void log_softmax_rows(float* __restrict__ x, int M, int C) {
    int m = blockIdx.x * blockDim.x + threadIdx.x;
    if (m >= M) return;
    float* row = x + (size_t)m * C;
    float mx = -3.4e38f;
    for (int c = 0; c < C; c++) mx = fmaxf(mx, row[c]);
    float s = 0.0f;
    for (int c = 0; c < C; c++) s += expf(row[c] - mx);
    float l = logf(s);
    for (int c = 0; c < C; c++) row[c] = row[c] - mx - l;
}

static inline void launch_gemm(const float* A, const float* B, const float* bias, float* C,
                               int M, int Nc, int K, int relu, hipStream_t stream) {
    int tiles = cdiv_h(M, 16) * cdiv_h(Nc, 16);
    gemm_wmma<<<cdiv_h(tiles, 4), 128, 0, stream>>>(A, B, bias, C, M, Nc, K, relu);
}

extern "C" void kernel_launch(void* const* d_in, const int* in_sizes, int n_in,
                              void* d_out, int out_size, void* d_ws, size_t ws_size,
                              hipStream_t stream) {
    const float* pos    = (const float*)d_in[0];
    // d_in[1] = batch (all zeros, unused)
    const float* kp1    = (const float*)d_in[2];
    const float* W1     = (const float*)d_in[3];   // (15,3,32)  -> (45,32)
    const float* kp2    = (const float*)d_in[4];
    const float* W2     = (const float*)d_in[5];   // (15,32,64) -> (480,64)
    const float* w_fp2  = (const float*)d_in[6];
    const float* b_fp2  = (const float*)d_in[7];
    const float* w_fp1a = (const float*)d_in[8];
    const float* b_fp1a = (const float*)d_in[9];
    const float* w_fp1b = (const float*)d_in[10];
    const float* b_fp1b = (const float*)d_in[11];
    const float* w_c1   = (const float*)d_in[12];
    const float* b_c1   = (const float*)d_in[13];
    const float* w_c2   = (const float*)d_in[14];
    const float* b_c2   = (const float*)d_in[15];
    float* out = (float*)d_out;

    float* W = (float*)d_ws;
    size_t o = 0;
    float* p    = W + o; o += 3 * (size_t)N_PTS;
    float* red  = W + o; o += 130;
    int*   idx1 = (int*)(W + o); o += (size_t)M1C * KNN;
    float* d2_1 = W + o; o += (size_t)M1C * KNN;
    float* g1   = W + o; o += (size_t)M1C * KPTS * 3;     // (M1, 45)
    float* x1   = W + o; o += (size_t)M1C * 32;
    int*   idx2 = (int*)(W + o); o += (size_t)M2C * KNN;
    float* d2_2 = W + o; o += (size_t)M2C * KNN;
    float* g2   = W + o; o += (size_t)M2C * KPTS * 32;    // (M2, 480)
    float* x2   = W + o; o += (size_t)M2C * 64;
    float* u    = W + o; o += (size_t)M1C * 96;
    float* hu   = W + o; o += (size_t)M1C * 32;
    float* va   = W + o; o += (size_t)N_PTS * 32;
    float* vb   = W + o; o += (size_t)N_PTS * 32;
    float* v16  = W + o; o += (size_t)N_PTS * 16;

    // normalize positions
    minmax_partial<<<64, 256, 0, stream>>>(pos, 3 * N_PTS, red);
    minmax_final<<<1, 32, 0, stream>>>(red);
    normalize_k<<<cdiv_h(3 * N_PTS, 256), 256, 0, stream>>>(pos, red, p, 3 * N_PTS);

    // ---- KPConv layer 1: x1 (M1,32), feat = p itself ----
    knn_kernel<KNN><<<cdiv_h(M1C, 128), 128, 0, stream>>>(p, M1C, p, N_PTS, idx1, d2_1);
    kpconv_g<<<cdiv_h(M1C, 8), 256, 0, stream>>>(p, M1C, p, p, 3, idx1, d2_1, kp1, RAD1, g1);
    launch_gemm(g1, W1, nullptr, x1, M1C, 32, 45, 0, stream);

    // ---- KPConv layer 2: x2 (M2,64), feat = x1 over p1 ----
    knn_kernel<KNN><<<cdiv_h(M2C, 128), 128, 0, stream>>>(p, M2C, p, M1C, idx2, d2_2);
    kpconv_g<<<cdiv_h(M2C, 8), 256, 0, stream>>>(p, M2C, p, x1, 32, idx2, d2_2, kp2, RAD2, g2);
    launch_gemm(g2, W2, nullptr, x2, M2C, 64, 480, 0, stream);

    // ---- decoder: interp x2 -> p1, concat x1, FP2 MLP ----
    knn_interp3<<<cdiv_h(M1C, 128), 128, 0, stream>>>(p, M1C, p, M2C, x2, 64, u, 96, 0);
    copy_block<<<cdiv_h(M1C * 32, 256), 256, 0, stream>>>(x1, u, M1C, 32, 96, 64);
    launch_gemm(u, w_fp2, b_fp2, hu, M1C, 32, 96, 1, stream);

    // ---- interp hu -> all points, head MLPs ----
    knn_interp3<<<cdiv_h(N_PTS, 128), 128, 0, stream>>>(p, N_PTS, p, M1C, hu, 32, va, 32, 0);
    launch_gemm(va, w_fp1a, b_fp1a, vb, N_PTS, 32, 32, 1, stream);
    launch_gemm(vb, w_fp1b, b_fp1b, va, N_PTS, 32, 32, 1, stream);
    launch_gemm(va, w_c1, b_c1, v16, N_PTS, 16, 32, 1, stream);
    launch_gemm(v16, w_c2, b_c2, out, N_PTS, NCLS, 16, 0, stream);
    log_softmax_rows<<<cdiv_h(N_PTS, 256), 256, 0, stream>>>(out, N_PTS, NCLS);
}